// GCNGRU_81458349736533
// MI455X (gfx1250) — compile-verified
//
#include <hip/hip_runtime.h>
#include <hip/hip_bf16.h>

// Problem constants (from reference)
#define TT   16384
#define NN   16
#define CIN  2
#define SS   13      // M*ORDER + ORDER + 1

typedef __attribute__((ext_vector_type(2))) float v2f;
typedef __attribute__((ext_vector_type(8))) float v8f;

// ---------------------------------------------------------------------------
// Kernel 0: build normalized Laplacians, Lsum = sum_l Lhat_l, M_l = Lhat_l @ Lsum
// Lmats layout (floats): [0..255]=Lsum, [256..511]=M0, [512..767]=M1, [768..1023]=M2
// ---------------------------------------------------------------------------
__global__ __launch_bounds__(256) void gcngru_prep(const float* __restrict__ a,
                                                   float* __restrict__ Lmats) {
    __shared__ float s_d[48];
    __shared__ float s_dis[48];
    __shared__ float s_L[768];   // 3 x 16 x 16
    __shared__ float s_S[256];   // Lsum
    const int tid = threadIdx.x;

    if (tid < 48) {                       // d[l][j] = sum_i a[l][i][j]
        const int l = tid >> 4, j = tid & 15;
        float s = 0.f;
        for (int i = 0; i < 16; ++i) s += a[l * 256 + i * 16 + j];
        s_d[tid]   = s;
        s_dis[tid] = rsqrtf(s);
    }
    __syncthreads();

    for (int e = tid; e < 768; e += 256) {  // Lhat = dis_i * (diag(d) - a) * dis_j
        const int l = e >> 8, i = (e >> 4) & 15, j = e & 15;
        const float lij = ((i == j) ? s_d[l * 16 + i] : 0.f) - a[e];
        s_L[e] = s_dis[l * 16 + i] * lij * s_dis[l * 16 + j];
    }
    __syncthreads();

    {                                       // Lsum
        const float s = s_L[tid] + s_L[256 + tid] + s_L[512 + tid];
        s_S[tid]   = s;
        Lmats[tid] = s;
    }
    __syncthreads();

    for (int e = tid; e < 768; e += 256) {  // M_l = Lhat_l @ Lsum
        const int l = e >> 8, i = (e >> 4) & 15, j = e & 15;
        float acc = 0.f;
        for (int k = 0; k < 16; ++k)
            acc = fmaf(s_L[l * 256 + i * 16 + k], s_S[k * 16 + j], acc);
        Lmats[256 + e] = acc;
    }
}

// ---------------------------------------------------------------------------
// Kernel 1: xg[t][n] = relu(bx + sum_c WS_c @ x[:,c]) for all t, via f32 WMMA.
// Columns of X are (t,c) pairs, col = 2t + c; one wave handles a 16-column tile.
// Y = Lsum@X, Z_l = M_l@X computed with V_WMMA_F32_16X16X4_F32 (4 K-chunks each).
// ---------------------------------------------------------------------------
__device__ __forceinline__ v8f wmma4(v2f a, v2f b, v8f c) {
    // (neg_a, A, neg_b, B, c_mod, C, reuse_a, reuse_b)
    return __builtin_amdgcn_wmma_f32_16x16x4_f32(false, a, false, b, (short)0, c,
                                                 false, false);
}

__global__ __launch_bounds__(256) void gcngru_xg(const float* __restrict__ in,
                                                 const float* __restrict__ wx,
                                                 const float* __restrict__ bx,
                                                 const float* __restrict__ Lmats,
                                                 float* __restrict__ xg) {
    const int lane = threadIdx.x & 31;
    const int wave = (blockIdx.x * blockDim.x + threadIdx.x) >> 5;
    const int col0 = wave * 16;              // first (t,c) column of this tile
    const int jc   = lane & 15;              // row (A) / column (B,D) index for this lane
    const bool hi  = lane >= 16;

    // A operands: row-major 16x16 -> WMMA A layout, 4 chunks of K=4 each.
    // chunk i: lane<16 holds {W[r][4i+0], W[r][4i+1]}, lane>=16 holds {W[r][4i+2], W[r][4i+3]}
    v2f A[4][4];
#pragma unroll
    for (int m = 0; m < 4; ++m) {
        const float* W = Lmats + m * 256;
#pragma unroll
        for (int i = 0; i < 4; ++i) {
            const int k0 = 4 * i + (hi ? 2 : 0);
            A[m][i].x = W[jc * 16 + k0];
            A[m][i].y = W[jc * 16 + k0 + 1];
        }
    }

    // B operand: X tile. X[m][col] = in[t*32 + m*2 + c], col = col0 + jc.
    const int col = col0 + jc;
    const int t   = col >> 1;
    const int c   = col & 1;
    v2f Bx[4];
#pragma unroll
    for (int i = 0; i < 4; ++i) {
        const int m0 = 4 * i + (hi ? 2 : 0);
        Bx[i].x = in[t * 32 + m0 * 2 + c];
        Bx[i].y = in[t * 32 + (m0 + 1) * 2 + c];
    }

    v8f Y = {}, Z0 = {}, Z1 = {}, Z2 = {};
#pragma unroll
    for (int i = 0; i < 4; ++i) {
        Y  = wmma4(A[0][i], Bx[i], Y);
        Z0 = wmma4(A[1][i], Bx[i], Z0);
        Z1 = wmma4(A[2][i], Bx[i], Z1);
        Z2 = wmma4(A[3][i], Bx[i], Z2);
    }

    // Per-column scalar mixing: WS_c@xcol = w10*x + w11*w0*Y + w12*w0*(w0*Z0+w1*Z1+w2*Z2)
    const float* w = wx + t * (CIN * SS) + c * SS;
    const float w0 = w[0], w1 = w[1], w2 = w[2];
    const float w10 = w[10], w11 = w[11], w12 = w[12];
    const float cA  = w11 * w0;
    const float cB0 = w12 * w0 * w0, cB1 = w12 * w0 * w1, cB2 = w12 * w0 * w2;

    // D layout: lane<16 holds rows 0..7 of column jc, lane>=16 rows 8..15.
    const int mbase = hi ? 8 : 0;
    float res[8];
#pragma unroll
    for (int r = 0; r < 8; ++r) {
        const float xv = in[t * 32 + (mbase + r) * 2 + c];
        res[r] = w10 * xv + cA * Y[r] + cB0 * Z0[r] + cB1 * Z1[r] + cB2 * Z2[r];
    }
    // Sum the two channels (columns 2t and 2t+1 live in lanes differing in bit 0)
#pragma unroll
    for (int r = 0; r < 8; ++r) res[r] += __shfl_xor(res[r], 1, 32);

    if ((lane & 1) == 0) {
#pragma unroll
        for (int r = 0; r < 8; ++r) {
            const float v = res[r] + bx[t * 16 + mbase + r];
            xg[t * 16 + mbase + r] = v > 0.f ? v : 0.f;
        }
    }
}

// ---------------------------------------------------------------------------
// Kernel 2: sequential GRU scan, one wave32. h component per lane, Lsum/M_l
// columns register-cached, shfl matvecs, prefetch next step's gru_k (6 KB).
// ---------------------------------------------------------------------------
__global__ __launch_bounds__(32) void gcngru_scan(const float* __restrict__ Lmats,
                                                  const float* __restrict__ xg,
                                                  const float* __restrict__ wh,
                                                  const float* __restrict__ bh,
                                                  const float* __restrict__ K,
                                                  const float* __restrict__ Bb,
                                                  float* __restrict__ out) {
    const int lane = threadIdx.x & 31;
    const int idx  = lane & 15;

    float Ls[16], M0c[16], M1c[16], M2c[16];
#pragma unroll
    for (int m = 0; m < 16; ++m) {
        Ls[m]  = Lmats[m * 16 + idx];
        M0c[m] = Lmats[256 + m * 16 + idx];
        M1c[m] = Lmats[512 + m * 16 + idx];
        M2c[m] = Lmats[768 + m * 16 + idx];
    }

    float h = 0.f;
    for (int t = 0; t < TT; ++t) {
        if (t + 1 < TT) {   // hide gru_k latency behind this step's dependent chain
            const char* pk = (const char*)(K + (size_t)(t + 1) * 1536);
            __builtin_prefetch(pk + lane * 128, 0, 1);            // first 4 KB
            __builtin_prefetch(pk + 4096 + lane * 64, 0, 1);      // remaining 2 KB
        }

        float hg;
        if (t == 0) {
            hg = 0.f;       // reference feeds literal zeros into the first GRU
        } else {
            const float* w = wh + t * SS;
            const float w0 = w[0], w1 = w[1], w2 = w[2];
            const float w10 = w[10], w11 = w[11], w12 = w[12];
            float y = 0.f, z0 = 0.f, z1 = 0.f, z2 = 0.f;
#pragma unroll
            for (int m = 0; m < 16; ++m) {
                const float hm = __shfl(h, m, 32);
                y  = fmaf(Ls[m],  hm, y);
                z0 = fmaf(M0c[m], hm, z0);
                z1 = fmaf(M1c[m], hm, z1);
                z2 = fmaf(M2c[m], hm, z2);
            }
            const float o = w10 * h + w11 * w0 * y +
                            w12 * w0 * (w0 * z0 + w1 * z1 + w2 * z2) + bh[t * 16 + idx];
            hg = o > 0.f ? o : 0.f;
        }

        const float  xv = xg[t * 16 + idx];
        const float* Kt = K + (size_t)t * 1536;
        const float* Bt = Bb + (size_t)t * 96;

        float a0 = 0.f, a1 = 0.f, a2 = 0.f, a3 = 0.f, a4 = 0.f;
#pragma unroll
        for (int m = 0; m < 16; ++m) {
            const float xm = __shfl(xv, m, 32);
            const float gm = __shfl(hg, m, 32);
            a0 = fmaf(Kt[        m * 16 + idx], xm, a0);
            a1 = fmaf(Kt[256  +  m * 16 + idx], gm, a1);
            a2 = fmaf(Kt[512  +  m * 16 + idx], xm, a2);
            a3 = fmaf(Kt[768  +  m * 16 + idx], gm, a3);
            a4 = fmaf(Kt[1024 +  m * 16 + idx], xm, a4);
        }
        const float r = 1.f / (1.f + __expf(-(a0 + Bt[idx] + a1 + Bt[16 + idx])));
        const float z = 1.f / (1.f + __expf(-(a2 + Bt[32 + idx] + a3 + Bt[48 + idx])));

        const float rh = r * hg;
        float a5 = 0.f;
#pragma unroll
        for (int m = 0; m < 16; ++m)
            a5 = fmaf(Kt[1280 + m * 16 + idx], __shfl(rh, m, 32), a5);

        const float hc = tanhf(a4 + Bt[64 + idx] + a5 + Bt[80 + idx]);
        h = z * hg + (1.f - z) * hc;

        if (lane < 16) out[t * 16 + lane] = h;
    }
}

// ---------------------------------------------------------------------------
extern "C" void kernel_launch(void* const* d_in, const int* in_sizes, int n_in,
                              void* d_out, int out_size, void* d_ws, size_t ws_size,
                              hipStream_t stream) {
    const float* inputs = (const float*)d_in[0];  // (T, N, C_IN)
    const float* a_list = (const float*)d_in[1];  // (M, N, N)
    const float* wx     = (const float*)d_in[2];  // (T, C_IN, S)
    const float* bx     = (const float*)d_in[3];  // (T, N)
    const float* wh     = (const float*)d_in[4];  // (T, 1, S)
    const float* bh     = (const float*)d_in[5];  // (T, N)
    const float* gk     = (const float*)d_in[6];  // (T, 6, N, N)
    const float* gb     = (const float*)d_in[7];  // (T, 6, N)
    float* out = (float*)d_out;                   // (T, N)

    float* Lmats = (float*)d_ws;                  // 1024 floats
    float* xg    = Lmats + 1024;                  // T*N floats

    gcngru_prep<<<1, 256, 0, stream>>>(a_list, Lmats);

    // T*C_IN = 32768 columns -> 2048 waves of 16 columns; 8 waves per 256-thread block
    gcngru_xg<<<256, 256, 0, stream>>>(inputs, wx, bx, Lmats, xg);

    gcngru_scan<<<1, 32, 0, stream>>>(Lmats, xg, wh, bh, gk, gb, out);
}